// WarpingLayer_Flow_13572096656092
// MI455X (gfx1250) — compile-verified
//
#include <hip/hip_runtime.h>

// Problem constants (from setup_inputs): x [B,C,H,W], flow [B,2,H,W]
#define B_ 4
#define C_ 128
#define H_ 256
#define W_ 512
#define HW_ (H_ * W_)          // 131072
#define LOG2_W 9               // W = 512
#define LOG2_HW 17             // HW = 131072

__global__ __launch_bounds__(256) void warp_flow_kernel(
    const float* __restrict__ x,
    const float* __restrict__ flow,
    float* __restrict__ out)
{
    // one thread per output pixel (b,h,w); loop over C in registers
    const int p  = blockIdx.x * 256 + threadIdx.x;   // 0 .. B*H*W-1
    const int w  = p & (W_ - 1);
    const int hw = p & (HW_ - 1);
    const int h  = hw >> LOG2_W;
    const int b  = p >> LOG2_HW;

    // ---- per-pixel coordinate math (done once, amortized over 128 channels) ----
    const float flow_x = flow[(size_t)b * 2 * HW_ + hw];
    const float flow_y = flow[(size_t)b * 2 * HW_ + HW_ + hw];

    // mirror the reference arithmetic order:
    // gx = linspace(-1,1,W)[w] + flow_x * 2/(W-1);  fx = (gx+1)*0.5*(W-1)
    const float stepx = 2.0f / (float)(W_ - 1);
    const float stepy = 2.0f / (float)(H_ - 1);
    const float gx = (-1.0f + (float)w * stepx) + flow_x * stepx;
    const float gy = (-1.0f + (float)h * stepy) + flow_y * stepy;
    const float fx = (gx + 1.0f) * 0.5f * (float)(W_ - 1);
    const float fy = (gy + 1.0f) * 0.5f * (float)(H_ - 1);

    const float x0f = floorf(fx);
    const float y0f = floorf(fy);
    const float x1f = x0f + 1.0f;
    const float y1f = y0f + 1.0f;

    const float wx1 = fx - x0f, wx0 = 1.0f - wx1;
    const float wy1 = fy - y0f, wy0 = 1.0f - wy1;

    const float vx0 = (x0f >= 0.0f && x0f <= (float)(W_ - 1)) ? 1.0f : 0.0f;
    const float vx1 = (x1f >= 0.0f && x1f <= (float)(W_ - 1)) ? 1.0f : 0.0f;
    const float vy0 = (y0f >= 0.0f && y0f <= (float)(H_ - 1)) ? 1.0f : 0.0f;
    const float vy1 = (y1f >= 0.0f && y1f <= (float)(H_ - 1)) ? 1.0f : 0.0f;

    // zero-padding: weight * validity (corner order matches reference loop)
    float w00 = (wx0 * wy0) * (vx0 * vy0);
    float w10 = (wx1 * wy0) * (vx1 * vy0);
    float w01 = (wx0 * wy1) * (vx0 * vy1);
    float w11 = (wx1 * wy1) * (vx1 * vy1);

    const float mask_val = ((w00 + w10) + w01) + w11;
    const float mask = (mask_val >= 0.9999f) ? 1.0f : 0.0f;
    // mask is exactly 0 or 1 -> folding into weights is bit-identical to out*mask
    w00 *= mask; w10 *= mask; w01 *= mask; w11 *= mask;

    // clamped gather indices (weights already zero OOB corners)
    const int xi0 = (int)fminf(fmaxf(x0f, 0.0f), (float)(W_ - 1));
    const int xi1 = (int)fminf(fmaxf(x1f, 0.0f), (float)(W_ - 1));
    const int yi0 = (int)fminf(fmaxf(y0f, 0.0f), (float)(H_ - 1));
    const int yi1 = (int)fminf(fmaxf(y1f, 0.0f), (float)(H_ - 1));

    const int o00 = yi0 * W_ + xi0;
    const int o10 = yi0 * W_ + xi1;
    const int o01 = yi1 * W_ + xi0;
    const int o11 = yi1 * W_ + xi1;

    const float* __restrict__ xb  = x   + (size_t)b * C_ * HW_;
    float* __restrict__       ob  = out + (size_t)b * C_ * HW_ + hw;

    // ---- channel loop: 4 gathers + blend + streaming store per channel ----
    // unroll x4 -> 16 loads in flight per thread (MLP), stores are NT so the
    // 256MB output stream doesn't evict reusable x lines from the 192MB L2.
#pragma unroll 4
    for (int c = 0; c < C_; ++c) {
        const float* __restrict__ pc = xb + (size_t)c * HW_;
        const float v00 = pc[o00];
        const float v10 = pc[o10];
        const float v01 = pc[o01];
        const float v11 = pc[o11];
        const float v = ((w00 * v00 + w10 * v10) + w01 * v01) + w11 * v11;
        __builtin_nontemporal_store(v, ob + (size_t)c * HW_);
    }
}

extern "C" void kernel_launch(void* const* d_in, const int* in_sizes, int n_in,
                              void* d_out, int out_size, void* d_ws, size_t ws_size,
                              hipStream_t stream)
{
    const float* x    = (const float*)d_in[0];   // [4,128,256,512] fp32
    const float* flow = (const float*)d_in[1];   // [4,2,256,512]   fp32
    float* out        = (float*)d_out;           // [4,128,256,512] fp32

    const int pixels = B_ * H_ * W_;             // 524288
    const int block  = 256;                      // 8 wave32 waves
    const int grid   = pixels / block;           // 2048 blocks

    warp_flow_kernel<<<grid, block, 0, stream>>>(x, flow, out);
}